// _CommonConv1d_58660663329278
// MI455X (gfx1250) — compile-verified
//
#include <hip/hip_runtime.h>

// Problem constants (match reference)
#define BATCH   32
#define C_IN    64
#define C_OUT   128
#define LEN     16384
#define KS      7          // taps, padded to 8 for two 16x16x4 f32 WMMA steps

typedef __attribute__((ext_vector_type(2))) float v2f;
typedef __attribute__((ext_vector_type(8))) float v8f;

// ---------------------------------------------------------------------------
// Kernel 1: channel reduction  s[b, l] = sum_c x[b, c, l]
// float4-vectorized: 128-bit coalesced loads/stores. Reads 134 MB (the big
// input pass), writes 2 MB intermediate that stays hot in the 192 MB L2.
// ---------------------------------------------------------------------------
__global__ __launch_bounds__(256) void
reduce_channels(const float* __restrict__ x, float* __restrict__ s) {
    const int idx = blockIdx.x * 256 + threadIdx.x;   // one thread per 4 l's
    const int b   = idx / (LEN / 4);
    const int l4  = idx % (LEN / 4);

    const float4* xb = (const float4*)(x + (size_t)b * C_IN * LEN) + l4;
    float4 acc = make_float4(0.f, 0.f, 0.f, 0.f);
#pragma unroll 8
    for (int c = 0; c < C_IN; ++c) {
        float4 v = xb[c * (LEN / 4)];
        acc.x += v.x; acc.y += v.y; acc.z += v.z; acc.w += v.w;
    }
    ((float4*)(s + (size_t)b * LEN))[l4] = acc;
}

// ---------------------------------------------------------------------------
// Kernel 2: out[b, o, l] = sum_k w[o,k] * s[b, l+k-3] + bias[o]
// via V_WMMA_F32_16X16X4_F32:  D(16x16) = A(16x4) x B(4x16) + C
//   M = 16 output channels, N = 16 positions, K = taps (7 padded to 8 -> 2 WMMA)
//   C preloaded with bias (folded into the MAC chain).
// Block = 256 threads = 8 wave32; each wave owns a 16-wide l tile; block
// covers 128 consecutive l. grid = (LEN/128, BATCH).
// ---------------------------------------------------------------------------
__device__ __forceinline__ float ld_clamp(const float* __restrict__ p, int i) {
    return (i >= 0 && i < LEN) ? p[i] : 0.0f;   // zero-padding halo
}

__global__ __launch_bounds__(256) void
conv_wmma(const float* __restrict__ s, const float* __restrict__ w,
          const float* __restrict__ bias, float* __restrict__ out) {
    // Stage weights (K padded to 8, tap 7 == 0) and bias in LDS.
    __shared__ float lds_w[C_OUT * 8];
    __shared__ float lds_b[C_OUT];

    const int tid = threadIdx.x;
    for (int i = tid; i < C_OUT * 8; i += 256) {
        const int o = i >> 3, k = i & 7;
        lds_w[i] = (k < KS) ? w[o * KS + k] : 0.0f;
    }
    for (int i = tid; i < C_OUT; i += 256) lds_b[i] = bias[i];
    __syncthreads();

    const int wave = tid >> 5;          // 0..7
    const int lane = tid & 31;
    const int half = lane >> 4;         // 0: lanes 0-15, 1: lanes 16-31
    const int n    = lane & 15;         // N (position) / M (channel) sub-index
    const int b    = blockIdx.y;
    const int lb   = blockIdx.x * 128 + wave * 16;   // l base of this wave

    const float* __restrict__ sb = s + (size_t)b * LEN;

    // B operand (4x16 f32, shared by all 8 M tiles):
    //   lane = N; lanes 0-15 hold K = {0,1} in the VGPR pair, lanes 16-31
    //   hold K = {2,3} (symmetric with the documented 16x4 A layout).
    //   B[k][n] = s[lb + n + k - 3]   (zero-padded at the ends)
    v2f bm1, bm2;
    {
        const int k0 = 2 * half;            // step 1: taps 0..3
        const int i0 = lb + n + k0 - 3;
        bm1.x = ld_clamp(sb, i0);
        bm1.y = ld_clamp(sb, i0 + 1);
        const int k4 = 4 + 2 * half;        // step 2: taps 4..7 (tap 7 -> w==0)
        const int j0 = lb + n + k4 - 3;
        bm2.x = ld_clamp(sb, j0);
        bm2.y = ld_clamp(sb, j0 + 1);
    }

    float* __restrict__ ob = out + (size_t)b * C_OUT * LEN + lb;

#pragma unroll
    for (int mt = 0; mt < 8; ++mt) {        // 8 tiles of 16 output channels
        const int mb = mt * 16;

        // A operand (16x4 f32): lane = M (= lane&15); lanes 0-15 hold K{0,1},
        // lanes 16-31 hold K{2,3}.
        const int ka = 2 * half;
        const float* wr = &lds_w[(mb + n) * 8];
        v2f a1, a2;
        a1.x = wr[ka];
        a1.y = wr[ka + 1];
        a2.x = wr[4 + ka];
        a2.y = wr[4 + ka + 1];              // K=7 slot is 0.0f -> padding tap

        // C/D (16x16 f32, 8 VGPRs): VGPR v -> M = v + 8*half, N = lane&15.
        // Initialize with bias so it rides the accumulation chain.
        v8f c;
#pragma unroll
        for (int v = 0; v < 8; ++v) c[v] = lds_b[mb + v + 8 * half];

        // D = A2 x B2 + (A1 x B1 + bias)   -- two chained K=4 steps
        c = __builtin_amdgcn_wmma_f32_16x16x4_f32(
                /*neg_a=*/false, a1, /*neg_b=*/false, bm1,
                /*c_mod=*/(short)0, c, /*reuse_a=*/false, /*reuse_b=*/false);
        c = __builtin_amdgcn_wmma_f32_16x16x4_f32(
                false, a2, false, bm2, (short)0, c, false, false);

        // Store: out[b, mb + v + 8*half, lb + n]; 16 lanes x 4 B = 64 B
        // contiguous per channel row, adjacent waves fill adjacent 64 B runs.
#pragma unroll
        for (int v = 0; v < 8; ++v)
            ob[(size_t)(mb + v + 8 * half) * LEN + n] = c[v];
    }
}

// ---------------------------------------------------------------------------
extern "C" void kernel_launch(void* const* d_in, const int* in_sizes, int n_in,
                              void* d_out, int out_size, void* d_ws, size_t ws_size,
                              hipStream_t stream) {
    const float* x    = (const float*)d_in[0];   // [32, 64, 16384]
    const float* wgt  = (const float*)d_in[1];   // [128, 1, 7]
    const float* bias = (const float*)d_in[2];   // [128]
    float*       out  = (float*)d_out;           // [32, 128, 16384]
    float*       s    = (float*)d_ws;            // [32, 16384] intermediate (2 MB)

    // Phase 1: channel sum (reads 134 MB, writes 2 MB -> L2 resident)
    const int n_thr = BATCH * (LEN / 4);         // 131072
    reduce_channels<<<n_thr / 256, 256, 0, stream>>>(x, s);

    // Phase 2: WMMA conv + bias (reads 2 MB from L2, writes 268 MB)
    conv_wmma<<<dim3(LEN / 128, BATCH), 256, 0, stream>>>(s, wgt, bias, out);
}